// SpeciesAwareRadialBasis_18485539242258
// MI455X (gfx1250) — compile-verified
//
#include <hip/hip_runtime.h>
#include <math.h>

typedef float v2f __attribute__((ext_vector_type(2)));
typedef float v8f __attribute__((ext_vector_type(8)));

#define NELEM 83
#define NEMB  64
#define NRAD  8

// ---------------------------------------------------------------------------
// Kernel 1: T[z][f] = sum_e embed_table[z][e] * dense_w[e][f] + bias[f]
// 83x64 @ 64x8 via V_WMMA_F32_16X16X4_F32. 6 waves, one 16-row M-tile each,
// 16 K-steps of 4, N=16 with columns >=8 zero-padded.
//
// Layouts (ISA 7.12.2, wave32):
//   A 16x4 : lane&15 = M, elem g / half h holds K = 2h + g
//   B 4x16 : lane&15 = N, elem g / half h holds K = 2h + g
//   D 16x16: lane&15 = N, elem g / half h holds M = g + 8h
// ---------------------------------------------------------------------------
__global__ __launch_bounds__(192) void precompute_species_table(
    const float* __restrict__ emb, const float* __restrict__ W,
    const float* __restrict__ bias, float* __restrict__ T) {
  const int tid  = threadIdx.x;
  const int wave = tid >> 5;          // 0..5  -> M tile
  const int lane = tid & 31;
  const int h    = lane >> 4;         // lane half
  const int l15  = lane & 15;
  const int rowBase = wave * 16;

  v8f acc = {};
  for (int ks = 0; ks < 16; ++ks) {
    const int k0  = ks * 4 + 2 * h;   // K covered by elem0; elem1 = k0+1
    const int row = rowBase + l15;    // A row (M)
    v2f a;
    if (row < NELEM) {
      a.x = emb[row * NEMB + k0];
      a.y = emb[row * NEMB + k0 + 1];
    } else { a.x = 0.0f; a.y = 0.0f; }
    v2f b;
    if (l15 < NRAD) {
      b.x = W[k0 * NRAD + l15];
      b.y = W[(k0 + 1) * NRAD + l15];
    } else { b.x = 0.0f; b.y = 0.0f; }
    acc = __builtin_amdgcn_wmma_f32_16x16x4_f32(
        /*neg_a=*/false, a, /*neg_b=*/false, b,
        /*c_mod=*/(short)0, acc, /*reuse_a=*/false, /*reuse_b=*/false);
  }

  if (l15 < NRAD) {
    const float bv = bias[l15];
#pragma unroll
    for (int g = 0; g < 8; ++g) {
      const int z = rowBase + g + 8 * h;   // D row (M)
      if (z < NELEM) T[z * NRAD + l15] = acc[g] + bv;
    }
  }
}

// ---------------------------------------------------------------------------
// Kernel 2: per-edge basis expansion. One thread computes one edge's staging
// values (16 sh + 8 rbf + 8 t) into LDS; each wave then streams its 32 edges
// out as coalesced 512B bursts (one b128 store per lane per edge).
// ---------------------------------------------------------------------------
__global__ __launch_bounds__(256) void edge_basis_kernel(
    const float* __restrict__ disp, const int* __restrict__ zj,
    const float* __restrict__ T, const float* __restrict__ tw,
    float* __restrict__ out, int ne) {
  __shared__ float stage[256][33];     // stride 33 -> bank-conflict free

  const int tid = threadIdx.x;
  const long long edge = (long long)blockIdx.x * 256 + tid;

  if (edge < ne) {
    float* s = stage[tid];
    const float x = disp[edge * 3 + 0];
    const float y = disp[edge * 3 + 1];
    const float z = disp[edge * 3 + 2];
    const float r   = sqrtf(x * x + y * y + z * z);
    const float inv = 1.0f / r;
    const float ux = x * inv, uy = y * inv, uz = z * inv;

    // spherical harmonics (Racah norm, spherical m order)
    const float s3 = 1.7320508075688772f, s6 = 2.449489742783178f;
    const float s10 = 3.1622776601683795f, s15 = 3.872983346207417f;
    const float xx = ux * ux, yy = uy * uy, zz = uz * uz;
    s[0] = 1.0f;
    s[1] = uy;  s[2] = uz;  s[3] = ux;
    s[4] = s3 * ux * uy;
    s[5] = s3 * uy * uz;
    s[6] = 0.5f * (3.0f * zz - 1.0f);
    s[7] = s3 * ux * uz;
    s[8] = 0.5f * s3 * (xx - yy);
    s[9]  = 0.25f * s10 * uy * (3.0f * xx - yy);
    s[10] = s15 * ux * uy * uz;
    s[11] = 0.25f * s6 * uy * (5.0f * zz - 1.0f);
    s[12] = 0.5f * uz * (5.0f * zz - 3.0f);
    s[13] = 0.25f * s6 * ux * (5.0f * zz - 1.0f);
    s[14] = 0.5f * s15 * uz * (xx - yy);
    s[15] = 0.25f * s10 * ux * (xx - 3.0f * yy);

    // jinc-like radial basis: sinc_k = sin(k*theta)/(k*theta), theta = pi*r/5
    // Chebyshev recurrence: sin((k+1)t) = 2cos(t) sin(kt) - sin((k-1)t)
    const float theta = 3.14159265358979323846f * r * 0.2f;
    float sn, cs;
    sincosf(theta, &sn, &cs);
    const float invth = 1.0f / theta;
    const float twoc  = 2.0f * cs;
    float sprev = 0.0f, scur = sn;
    float sinc[10];
#pragma unroll
    for (int k = 1; k <= 9; ++k) {
      sinc[k] = scur * invth * (1.0f / (float)k);
      const float snext = twoc * scur - sprev;
      sprev = scur; scur = snext;
    }
    const float factor1 = 0.3973835306318440f;  // sqrt(2)*pi / 5^1.5
    const float F2[8] = {  0.8944271909999159f, -1.6641005886756874f,
                           2.4000000000000000f, -3.1234752377721210f,
                           3.8411063979868794f, -4.5555402243460740f,
                           5.2680254627179970f, -5.9792740955759840f };
#pragma unroll
    for (int i = 0; i < 8; ++i)
      s[16 + i] = factor1 * F2[i] * (sinc[i + 1] + sinc[i + 2]);

    // species-dependent channel weights from precomputed (83,8) table
    const int Z = zj[edge];
    const float* tp = T + Z * NRAD;
#pragma unroll
    for (int f = 0; f < 8; ++f) s[24 + f] = tp[f];
  }
  __syncthreads();

  // phase 2: coalesced write. lane L owns (lm = L>>1, f = (L&1)*4 .. +3)
  const int wave = tid >> 5, lane = tid & 31;
  const int lm = lane >> 1;
  const int f0 = (lane & 1) * 4;
  const int l  = (lm >= 9) ? 3 : ((lm >= 4) ? 2 : ((lm >= 1) ? 1 : 0));
  const float w0 = tw[l * 8 + f0 + 0];
  const float w1 = tw[l * 8 + f0 + 1];
  const float w2 = tw[l * 8 + f0 + 2];
  const float w3 = tw[l * 8 + f0 + 3];
  const float res = (lm == 0) ? 1.0f : 0.0f;   // l=0 residual: y += t

  const long long blockBase = (long long)blockIdx.x * 256;
#pragma unroll 4
  for (int e = 0; e < 32; ++e) {
    const int row = wave * 32 + e;
    const long long eg = blockBase + row;
    if (eg >= ne) break;
    const float shv = stage[row][lm];
    const float rb0 = stage[row][16 + f0 + 0];
    const float rb1 = stage[row][16 + f0 + 1];
    const float rb2 = stage[row][16 + f0 + 2];
    const float rb3 = stage[row][16 + f0 + 3];
    const float t0  = stage[row][24 + f0 + 0];
    const float t1  = stage[row][24 + f0 + 1];
    const float t2  = stage[row][24 + f0 + 2];
    const float t3  = stage[row][24 + f0 + 3];
    float4 o;
    o.x = w0 * t0 * shv * rb0 + res * t0;
    o.y = w1 * t1 * shv * rb1 + res * t1;
    o.z = w2 * t2 * shv * rb2 + res * t2;
    o.w = w3 * t3 * shv * rb3 + res * t3;
    *(float4*)(out + eg * 128 + lane * 4) = o;   // wave: 512B contiguous burst
  }
}

// ---------------------------------------------------------------------------
extern "C" void kernel_launch(void* const* d_in, const int* in_sizes, int n_in,
                              void* d_out, int out_size, void* d_ws, size_t ws_size,
                              hipStream_t stream) {
  const float* disp = (const float*)d_in[0];   // (N, K, 3) f32
  const int*   zj   = (const int*)d_in[1];     // (N, K)    i32
  const float* emb  = (const float*)d_in[2];   // (83, 64)  f32
  const float* dw   = (const float*)d_in[3];   // (64, 8)   f32
  const float* db   = (const float*)d_in[4];   // (8,)      f32
  const float* tw   = (const float*)d_in[5];   // (4, 8)    f32
  float* out = (float*)d_out;
  float* T   = (float*)d_ws;                   // (83, 8) precomputed table

  precompute_species_table<<<1, 192, 0, stream>>>(emb, dw, db, T);

  const int ne = in_sizes[1];                  // N_ATOMS * N_NBR
  const int blocks = (ne + 255) / 256;
  edge_basis_kernel<<<blocks, 256, 0, stream>>>(disp, zj, T, tw, out, ne);
}